// DAAdj_57114475102829
// MI455X (gfx1250) — compile-verified
//
#include <hip/hip_runtime.h>
#include <hip/hip_bf16.h>

#define NN 4096
#define FF 256
#define HH 8

typedef __attribute__((ext_vector_type(2))) float v2f;
typedef __attribute__((ext_vector_type(8))) float v8f;

// ---------------------------------------------------------------------------
// Kernel 1: ST = x @ [Ws | Wt]^T  -> (N,16); cols 0..7 = s heads, 8..15 = t.
// Exact f32 GEMM via V_WMMA_F32_16X16X4_F32. One wave per 16-row tile;
// the full 16-wide output dimension fits one WMMA tile (8 s-heads + 8 t-heads).
// A 16x4 layout: lane<16 holds K=0,1 (vgpr0,1); lane>=16 holds K=2,3.
// B 4x16 layout mirrors A; D is the standard 16x16 f32 C/D layout.
// ---------------------------------------------------------------------------
__global__ void k_gemm_st(const float* __restrict__ x,
                          const float* __restrict__ wd,   // W_dist (8 x 512)
                          float* __restrict__ st) {
  const int lane = threadIdx.x & 31;
  const int wave = threadIdx.x >> 5;
  const int tile = blockIdx.x * 8 + wave;       // 16-row tile of x
  const int i0   = tile * 16;
  const int m    = lane & 15;                   // A row in tile == B column
  const int koff = (lane >> 4) * 2;             // 0 for lanes 0-15, 2 for 16-31
  const int h    = m & 7;                       // head
  const int part = m >> 3;                      // 0 -> Ws half, 1 -> Wt half

  const float* arow = x  + (size_t)(i0 + m) * FF;
  const float* brow = wd + (size_t)h * (2 * FF) + (size_t)part * FF;

  v8f acc = {};
#pragma unroll 4
  for (int kk = 0; kk < FF; kk += 4) {
    v2f a = *(const v2f*)(arow + kk + koff);
    v2f b = *(const v2f*)(brow + kk + koff);
    // 8 args: (neg_a, A, neg_b, B, c_mod, C, reuse_a, reuse_b)
    acc = __builtin_amdgcn_wmma_f32_16x16x4_f32(false, a, false, b,
                                                (short)0, acc, false, false);
  }

  const int rbase = i0 + ((lane < 16) ? 0 : 8);
#pragma unroll
  for (int v = 0; v < 8; ++v)
    st[(size_t)(rbase + v) * 16 + m] = acc[v];
}

// ---------------------------------------------------------------------------
// Kernel 2: P[i,h] = exp(s[i,h] + b_dist[h]);  Q[j,h] = exp(t[j,h]).
// Uses exp(relu(z)) == max(exp(z),1) factorization downstream, so these are
// the ONLY transcendentals (64K instead of 134M).
// ---------------------------------------------------------------------------
__global__ void k_exp_pq(const float* __restrict__ st,
                         const float* __restrict__ b_dist,
                         float* __restrict__ P,
                         float* __restrict__ Q) {
  int idx = blockIdx.x * blockDim.x + threadIdx.x;
  if (idx >= NN * HH) return;
  int i = idx >> 3, h = idx & 7;
  float s = st[(size_t)i * 16 + h];
  float t = st[(size_t)i * 16 + 8 + h];
  P[idx] = __expf(s + b_dist[h]);
  Q[idx] = __expf(t);
}

// ---------------------------------------------------------------------------
// Kernel 3: denom[i,h] = sum_j max(P[i,h]*Q[j,h], 1)  (+ diagonal fix),
//           C[i,h]     = W_merge[h] / denom[i,h].
// One block per row i; deterministic LDS tree reduction (no float atomics).
// ---------------------------------------------------------------------------
__global__ void k_denom(const float* __restrict__ P,
                        const float* __restrict__ Q,
                        const float* __restrict__ w_merge,
                        const float* __restrict__ selfbias,  // (H,1)
                        float* __restrict__ C) {
  __shared__ float red[256 * HH];
  const int i = blockIdx.x;
  const int tid = threadIdx.x;

  float p[HH];
#pragma unroll
  for (int h = 0; h < HH; ++h) p[h] = P[(size_t)i * HH + h];

  float acc[HH] = {};
  for (int j = tid; j < NN; j += 256) {
    const float* q = Q + (size_t)j * HH;
#pragma unroll
    for (int h = 0; h < HH; ++h)
      acc[h] += fmaxf(p[h] * q[h], 1.0f);
    if (j == i) {  // diagonal: dist gets +selfbias[h] before relu/exp
#pragma unroll
      for (int h = 0; h < HH; ++h) {
        float pq = p[h] * q[h];
        acc[h] += fmaxf(pq * __expf(selfbias[h]), 1.0f) - fmaxf(pq, 1.0f);
      }
    }
  }

#pragma unroll
  for (int h = 0; h < HH; ++h) red[tid * HH + h] = acc[h];
  __syncthreads();
  for (int stride = 128; stride > 0; stride >>= 1) {
    if (tid < stride) {
#pragma unroll
      for (int h = 0; h < HH; ++h)
        red[tid * HH + h] += red[(tid + stride) * HH + h];
    }
    __syncthreads();
  }
  if (tid < HH) C[(size_t)i * HH + tid] = w_merge[tid] / red[tid];
}

// ---------------------------------------------------------------------------
// Kernel 4: out[i,j] = b_merge + sum_h C[i,h] * max(P[i,h]*Q[j,h], 1).
// One block per row i; Q (128 KB) stays L2-resident across all 4096 rows.
// Coalesced 64 MiB output stream -> this kernel rides the HBM write floor.
// ---------------------------------------------------------------------------
__global__ void k_out(const float* __restrict__ P,
                      const float* __restrict__ Q,
                      const float* __restrict__ C,
                      const float* __restrict__ selfbias,
                      const float* __restrict__ b_merge,
                      float* __restrict__ out) {
  const int i = blockIdx.x;
  const int tid = threadIdx.x;

  float p[HH], c[HH];
#pragma unroll
  for (int h = 0; h < HH; ++h) {
    p[h] = P[(size_t)i * HH + h];
    c[h] = C[(size_t)i * HH + h];
  }
  const float bm = b_merge[0];

  for (int j0 = 0; j0 < NN; j0 += 256) {
    const int j = j0 + tid;
    const float* q = Q + (size_t)j * HH;
    float r = bm;
    if (j == i) {
#pragma unroll
      for (int h = 0; h < HH; ++h)
        r += c[h] * fmaxf(p[h] * q[h] * __expf(selfbias[h]), 1.0f);
    } else {
#pragma unroll
      for (int h = 0; h < HH; ++h)
        r += c[h] * fmaxf(p[h] * q[h], 1.0f);
    }
    out[(size_t)i * NN + j] = r;
  }
}

// ---------------------------------------------------------------------------
extern "C" void kernel_launch(void* const* d_in, const int* in_sizes, int n_in,
                              void* d_out, int out_size, void* d_ws, size_t ws_size,
                              hipStream_t stream) {
  const float* x  = (const float*)d_in[0];   // (N,F)
  const float* wd = (const float*)d_in[1];   // (H,2F)
  const float* bd = (const float*)d_in[2];   // (H,)
  const float* wm = (const float*)d_in[3];   // (1,H)
  const float* bm = (const float*)d_in[4];   // (1,)
  const float* sb = (const float*)d_in[5];   // (H,1)
  float* out = (float*)d_out;                // (N,N)

  float* ws = (float*)d_ws;
  float* st = ws;                            // N*16
  float* P  = st + (size_t)NN * 16;          // N*H
  float* Q  = P  + (size_t)NN * HH;          // N*H
  float* C  = Q  + (size_t)NN * HH;          // N*H
  (void)in_sizes; (void)n_in; (void)out_size; (void)ws_size;

  k_gemm_st<<<NN / 16 / 8, 256, 0, stream>>>(x, wd, st);
  k_exp_pq<<<(NN * HH + 255) / 256, 256, 0, stream>>>(st, bd, P, Q);
  k_denom<<<NN, 256, 0, stream>>>(P, Q, wm, sb, C);
  k_out<<<NN, 256, 0, stream>>>(P, Q, C, sb, bm, out);
}